// TripletLoss_79731772882978
// MI455X (gfx1250) — compile-verified
//
#include <hip/hip_runtime.h>
#include <hip/hip_bf16.h>
#include <math.h>

typedef float v2f __attribute__((ext_vector_type(2)));
typedef float v8f __attribute__((ext_vector_type(8)));

#define N_PTS   8192
#define DIM     128
#define MARGIN  0.5f
#define BIGF    1e30f
#define VALIDF  1e29f

// ---------------------------------------------------------------------------
// Kernel 1: squared row norms. One wave per row (wave32), float4 per lane.
// ---------------------------------------------------------------------------
__global__ void triplet_rownorm_kernel(const float* __restrict__ emb,
                                       float* __restrict__ sq) {
    const int wave = threadIdx.x >> 5;
    const int lane = threadIdx.x & 31;
    const int row  = blockIdx.x * 8 + wave;   // 256 thr = 8 waves = 8 rows
    const float4 v = *(const float4*)(emb + (size_t)row * DIM + lane * 4);
    float s = v.x * v.x + v.y * v.y + v.z * v.z + v.w * v.w;
    #pragma unroll
    for (int m = 16; m >= 1; m >>= 1)
        s += __shfl_xor(s, m, 32);
    if (lane == 0) sq[row] = s;
}

// ---------------------------------------------------------------------------
// Kernel 2: fused  G = X·X^T  (WMMA f32 16x16x4)  +  squared-distance
// batch-hard mining. Block = 256 threads = 8 waves = 2 row-tiles x 4
// col-tiles. Block owns rows [32*b, 32*b+32) exclusively -> deterministic.
// Mining is done on squared distances (sqrt is monotonic); sqrt applied once
// per row at the end.
// ---------------------------------------------------------------------------
__global__ void __launch_bounds__(256)
triplet_mine_kernel(const float* __restrict__ emb,
                    const int*   __restrict__ labels,
                    const float* __restrict__ sq,
                    float* __restrict__ rowloss,
                    float* __restrict__ rowvalid) {
    __shared__ float sp[128];   // [wave][16] partial hardest-positive (d^2)
    __shared__ float sn[128];   // [wave][16] partial hardest-negative (d^2)

    const int w    = threadIdx.x >> 5;
    const int lane = threadIdx.x & 31;
    const int rt   = w >> 2;          // row-tile 0..1
    const int ct   = w & 3;           // col-tile 0..3
    const int hi   = lane >> 4;       // half-wave
    const int l15  = lane & 15;
    const int koff = hi * 2;          // k-offset within 4-wide K step

    const int ibase = blockIdx.x * 32 + rt * 16;   // this wave's 16 rows

    // --- Preload A fragments for all 32 K-steps into registers (reused for
    //     every column tile). Lane layout per ISA 7.12.2 (32-bit A 16x4).
    const int rowA = ibase + l15;
    v2f areg[32];
    #pragma unroll
    for (int s = 0; s < 32; ++s)
        areg[s] = *(const v2f*)(emb + (size_t)rowA * DIM + 4 * s + koff);

    // Row-side constants for the epilogue (C layout: lane l, vgpr r ->
    // row = r + 8*hi, col = jt + (l&15)).
    float sqi[8];
    int   labi[8];
    #pragma unroll
    for (int r = 0; r < 8; ++r) {
        const int gi = ibase + r + 8 * hi;
        sqi[r]  = sq[gi];
        labi[r] = labels[gi];
    }

    float posm[8], negm[8];           // squared-distance extrema
    #pragma unroll
    for (int r = 0; r < 8; ++r) { posm[r] = -BIGF; negm[r] = BIGF; }

    // --- Sweep all columns; each iteration this wave does one 16x16 tile.
    for (int j0 = 0; j0 < N_PTS; j0 += 64) {
        const int jcol = j0 + ct * 16 + l15;           // this lane's column
        const float sqj = sq[jcol];
        const int   labj = labels[jcol];
        const int   rowB = jcol;

        // Two independent accumulator chains (even/odd K-steps) so the
        // matrix pipe can interleave two WMMA dependency chains.
        v8f c0 = {}, c1 = {};
        #pragma unroll
        for (int s = 0; s < 32; s += 2) {
            v2f b0 = *(const v2f*)(emb + (size_t)rowB * DIM + 4 * s + koff);
            v2f b1 = *(const v2f*)(emb + (size_t)rowB * DIM + 4 * (s + 1) + koff);
            c0 = __builtin_amdgcn_wmma_f32_16x16x4_f32(
                     false, areg[s],     false, b0, (short)0, c0, false, false);
            c1 = __builtin_amdgcn_wmma_f32_16x16x4_f32(
                     false, areg[s + 1], false, b1, (short)0, c1, false, false);
        }

        // Epilogue on squared distances (no sqrt in the hot loop).
        #pragma unroll
        for (int r = 0; r < 8; ++r) {
            const float g  = c0[r] + c1[r];
            float d2 = fmaf(-2.0f, g, sqi[r] + sqj);
            d2 = fmaxf(d2, 0.0f);
            const int gi = ibase + r + 8 * hi;
            if (labi[r] == labj) {
                if (gi != jcol) posm[r] = fmaxf(posm[r], d2);
            } else {
                negm[r] = fminf(negm[r], d2);
            }
        }
    }

    // --- Reduce across the 16 lanes of each half-wave (different columns,
    //     same rows).
    #pragma unroll
    for (int r = 0; r < 8; ++r) {
        #pragma unroll
        for (int m = 8; m >= 1; m >>= 1) {
            posm[r] = fmaxf(posm[r], __shfl_xor(posm[r], m, 16));
            negm[r] = fminf(negm[r], __shfl_xor(negm[r], m, 16));
        }
    }
    if (l15 == 0) {  // lanes 0 and 16 hold the half-wave results
        #pragma unroll
        for (int r = 0; r < 8; ++r) {
            const int idx = w * 16 + r + 8 * hi;
            sp[idx] = posm[r];
            sn[idx] = negm[r];
        }
    }
    __syncthreads();

    // --- Cross-wave combine (4 waves share each row-tile) + per-row loss.
    if (threadIdx.x < 32) {
        const int t   = threadIdx.x;
        const int rt2 = t >> 4;
        const int r16 = t & 15;
        float pos2 = -BIGF, neg2 = BIGF;
        #pragma unroll
        for (int k = 0; k < 4; ++k) {
            const int w2 = rt2 * 4 + k;
            pos2 = fmaxf(pos2, sp[w2 * 16 + r16]);
            neg2 = fminf(neg2, sn[w2 * 16 + r16]);
        }
        const int row = blockIdx.x * 32 + t;
        const bool valid = (pos2 > -VALIDF) && (neg2 < VALIDF);
        float loss = 0.0f;
        if (valid) {
            const float pos = sqrtf(fmaxf(pos2, 0.0f));
            const float neg = sqrtf(neg2);
            loss = fmaxf(pos - neg + MARGIN, 0.0f);
        }
        rowloss[row]  = loss;
        rowvalid[row] = valid ? 1.0f : 0.0f;
    }
}

// ---------------------------------------------------------------------------
// Kernel 3: deterministic final reduction, single block.
// ---------------------------------------------------------------------------
__global__ void triplet_reduce_kernel(const float* __restrict__ rowloss,
                                      const float* __restrict__ rowvalid,
                                      float* __restrict__ out) {
    __shared__ float ssum[256];
    __shared__ float scnt[256];
    float s = 0.0f, c = 0.0f;
    for (int i = threadIdx.x; i < N_PTS; i += 256) {
        s += rowloss[i];
        c += rowvalid[i];
    }
    ssum[threadIdx.x] = s;
    scnt[threadIdx.x] = c;
    __syncthreads();
    for (int stride = 128; stride > 0; stride >>= 1) {
        if (threadIdx.x < stride) {
            ssum[threadIdx.x] += ssum[threadIdx.x + stride];
            scnt[threadIdx.x] += scnt[threadIdx.x + stride];
        }
        __syncthreads();
    }
    if (threadIdx.x == 0)
        out[0] = ssum[0] / fmaxf(scnt[0], 1.0f);
}

// ---------------------------------------------------------------------------
extern "C" void kernel_launch(void* const* d_in, const int* in_sizes, int n_in,
                              void* d_out, int out_size, void* d_ws, size_t ws_size,
                              hipStream_t stream) {
    (void)in_sizes; (void)n_in; (void)out_size; (void)ws_size;
    const float* emb    = (const float*)d_in[0];
    const int*   labels = (const int*)d_in[1];
    float* out = (float*)d_out;

    float* ws       = (float*)d_ws;
    float* sq       = ws;                 // N floats
    float* rowloss  = ws + N_PTS;         // N floats
    float* rowvalid = ws + 2 * N_PTS;     // N floats

    triplet_rownorm_kernel<<<N_PTS / 8, 256, 0, stream>>>(emb, sq);
    triplet_mine_kernel<<<N_PTS / 32, 256, 0, stream>>>(emb, labels, sq,
                                                        rowloss, rowvalid);
    triplet_reduce_kernel<<<1, 256, 0, stream>>>(rowloss, rowvalid, out);
}